// RoIAlign_18691697672636
// MI455X (gfx1250) — compile-verified
//
#include <hip/hip_runtime.h>
#include <hip/hip_bf16.h>

#define AS_GLOBAL __attribute__((address_space(1)))
#define AS_LDS    __attribute__((address_space(3)))

typedef int v4i __attribute__((ext_vector_type(4)));

#ifndef __has_builtin
#define __has_builtin(x) 0
#endif

namespace {
constexpr int C = 256, H = 200, W = 200;
constexpr int PH = 7, PW = 7;
constexpr int NBINS = PH * PW;             // 49
constexpr int NSAMP = NBINS * 4;           // 196 (2x2 samples per bin)
constexpr int OUT_PER_ROI = C * NBINS;     // 12544 floats = 50176 B
constexpr float SCALE = 0.25f;
}

__device__ __forceinline__ float ld_off(const char* base, int byte_off) {
  return *(const float*)(base + byte_off);
}

__global__ __launch_bounds__(256) void roialign_fwd(
    const float* __restrict__ feat, const float* __restrict__ rois,
    float* __restrict__ out, int K)
{
  __shared__ int4   s_off[NSAMP];   // byte offsets of the 4 bilinear corners
  __shared__ float4 s_w[NSAMP];     // pre-masked, pre-averaged weights
  __shared__ __align__(16) float s_out[OUT_PER_ROI];

  const int k   = blockIdx.x;
  const int tid = threadIdx.x;

  // ---------------- phase 1: per-RoI bilinear sample table ----------------
  // rois row = [b, y1, x1, y2, x2]; reference permutes to torchvision order.
  const float rb = rois[k * 5 + 0];
  if (tid < NSAMP) {
    const float y1 = rois[k * 5 + 1] * SCALE;
    const float x1 = rois[k * 5 + 2] * SCALE;
    const float y2 = rois[k * 5 + 3] * SCALE;
    const float x2 = rois[k * 5 + 4] * SCALE;
    const float roi_w = fmaxf(x2 - x1, 1.0f);
    const float roi_h = fmaxf(y2 - y1, 1.0f);
    const float bin_w = roi_w * (1.0f / (float)PW);
    const float bin_h = roi_h * (1.0f / (float)PH);

    const int bin = tid >> 2, s = tid & 3;
    const int ph = bin / PW, pw = bin % PW;
    const int iy = s >> 1,   ix = s & 1;

    const float x = x1 + ((float)pw + ((float)ix + 0.5f) * 0.5f) * bin_w;
    const float y = y1 + ((float)ph + ((float)iy + 0.5f) * 0.5f) * bin_h;

    // x axis (matches _axis_prep, aligned=False edge handling)
    float vx = (x >= -1.0f && x <= (float)W) ? 1.0f : 0.0f;
    float cx = fmaxf(x, 0.0f);
    int   x0 = min((int)floorf(cx), W - 1);
    int   x1i = min(x0 + 1, W - 1);
    float lx = (x0 >= W - 1) ? 0.0f : (cx - (float)x0);
    float hx = (1.0f - lx) * vx;
    lx *= vx;
    // y axis
    float vy = (y >= -1.0f && y <= (float)H) ? 1.0f : 0.0f;
    float cy = fmaxf(y, 0.0f);
    int   y0 = min((int)floorf(cy), H - 1);
    int   y1i = min(y0 + 1, H - 1);
    float ly = (y0 >= H - 1) ? 0.0f : (cy - (float)y0);
    float hy = (1.0f - ly) * vy;
    ly *= vy;

    // byte offsets: removes the per-gather <<2 in the hot loop
    s_off[tid] = make_int4((y0 * W + x0) * 4, (y0 * W + x1i) * 4,
                           (y1i * W + x0) * 4, (y1i * W + x1i) * 4);
    const float q = 0.25f;  // 2x2 sample average folded into weights
    s_w[tid] = make_float4(hy * hx * q, hy * lx * q, ly * hx * q, ly * lx * q);
  }
  __syncthreads();

  // ---------------- phase 2: gather-accumulate, one channel per thread ----
  const int b = (int)rb;
  const char* plane =
      (const char*)(feat + ((size_t)(b * C + tid)) * (size_t)(H * W));

  #pragma unroll 2
  for (int j = 0; j < NBINS; ++j) {
    // 4 independent partial sums (one per sample) -> short dependency
    // chains, lets the compiler clause the 16 gathers and batch waits.
    float a0 = 0.0f, a1 = 0.0f, a2 = 0.0f, a3 = 0.0f;
    {
      const int4 o = s_off[j * 4 + 0]; const float4 w = s_w[j * 4 + 0];
      a0 = fmaf(w.x, ld_off(plane, o.x), a0);
      a0 = fmaf(w.y, ld_off(plane, o.y), a0);
      a0 = fmaf(w.z, ld_off(plane, o.z), a0);
      a0 = fmaf(w.w, ld_off(plane, o.w), a0);
    }
    {
      const int4 o = s_off[j * 4 + 1]; const float4 w = s_w[j * 4 + 1];
      a1 = fmaf(w.x, ld_off(plane, o.x), a1);
      a1 = fmaf(w.y, ld_off(plane, o.y), a1);
      a1 = fmaf(w.z, ld_off(plane, o.z), a1);
      a1 = fmaf(w.w, ld_off(plane, o.w), a1);
    }
    {
      const int4 o = s_off[j * 4 + 2]; const float4 w = s_w[j * 4 + 2];
      a2 = fmaf(w.x, ld_off(plane, o.x), a2);
      a2 = fmaf(w.y, ld_off(plane, o.y), a2);
      a2 = fmaf(w.z, ld_off(plane, o.z), a2);
      a2 = fmaf(w.w, ld_off(plane, o.w), a2);
    }
    {
      const int4 o = s_off[j * 4 + 3]; const float4 w = s_w[j * 4 + 3];
      a3 = fmaf(w.x, ld_off(plane, o.x), a3);
      a3 = fmaf(w.y, ld_off(plane, o.y), a3);
      a3 = fmaf(w.z, ld_off(plane, o.z), a3);
      a3 = fmaf(w.w, ld_off(plane, o.w), a3);
    }
    s_out[tid * NBINS + j] = (a0 + a1) + (a2 + a3);
  }
  __syncthreads();

  // ---------------- phase 3: async LDS->global drain (coalesced) ---------
  float* __restrict__ dst = out + (size_t)k * (size_t)OUT_PER_ROI;
#if __has_builtin(__builtin_amdgcn_global_store_async_from_lds_b128)
  {
    for (int i = tid * 4; i < OUT_PER_ROI; i += 256 * 4) {
      __builtin_amdgcn_global_store_async_from_lds_b128(
          (AS_GLOBAL v4i*)(dst + i), (AS_LDS v4i*)&s_out[i], 0, 0);
    }
#if __has_builtin(__builtin_amdgcn_s_wait_asynccnt)
    __builtin_amdgcn_s_wait_asynccnt(0);
#endif
  }
#else
  for (int i = tid * 4; i < OUT_PER_ROI; i += 256 * 4) {
    *(float4*)(dst + i) = *(const float4*)&s_out[i];
  }
#endif
}

extern "C" void kernel_launch(void* const* d_in, const int* in_sizes, int n_in,
                              void* d_out, int out_size, void* d_ws, size_t ws_size,
                              hipStream_t stream) {
  const float* feature = (const float*)d_in[0];  // (4, 256, 200, 200) fp32
  const float* rois    = (const float*)d_in[1];  // (K, 5) fp32
  float*       out     = (float*)d_out;          // (K, 256, 7, 7) fp32
  const int K = in_sizes[1] / 5;
  (void)d_ws; (void)ws_size; (void)n_in; (void)out_size;
  roialign_fwd<<<K, 256, 0, stream>>>(feature, rois, out, K);
}